// DetrLoss_85925115724559
// MI455X (gfx1250) — compile-verified
//
#include <hip/hip_runtime.h>

typedef __attribute__((ext_vector_type(2))) float v2f;
typedef __attribute__((ext_vector_type(8))) float v8f;

#define B_DIM 128
#define Q_DIM 900
#define N_DIM 300
#define C_DIM 92          // 23 * 4  -> exact K tiling for 16x16x4
#define KSTEPS 23
#define L1_W 5.0f
#define GIOU_W 2.0f
#define INVALID_COST 1000000.0f

__global__ __launch_bounds__(32) void detr_cost_wmma(
    const float* __restrict__ pred_boxes,   // [B,Q,4]
    const float* __restrict__ pred_cls,     // [B,Q,C]
    const float* __restrict__ gt_boxes,     // [B,N,4]
    const int*   __restrict__ gt_cls,       // [B,N]
    const unsigned char* __restrict__ gt_valid, // [B,N] (bool)
    float* __restrict__ out)                // [B,Q,N]
{
    const int lane  = threadIdx.x;      // 0..31
    const int rlane = lane & 15;
    const int hi    = lane >> 4;        // 0: lanes 0-15, 1: lanes 16-31

    const int q0 = blockIdx.x * 16;
    const int n0 = blockIdx.y * 16;
    const int b  = blockIdx.z;

    // ---------- softmax stats for row (q0 + rlane), lane pair shares work ----------
    int qrow = q0 + rlane;
    if (qrow >= Q_DIM) qrow = Q_DIM - 1;          // clamp (dup row, never stored)
    const float* logits = pred_cls + ((size_t)b * Q_DIM + qrow) * C_DIM;

    const int cbeg = hi * 46;                      // lane pair splits 92 classes
    float m = -3.0e38f;
    for (int c = cbeg; c < cbeg + 46; ++c) m = fmaxf(m, logits[c]);
    m = fmaxf(m, __shfl_xor(m, 16, 32));           // combine pair -> row max

    float s = 0.0f;
    for (int c = cbeg; c < cbeg + 46; ++c) s += __expf(logits[c] - m);
    s += __shfl_xor(s, 16, 32);                    // combine pair -> row sum
    const float inv_s = 1.0f / s;

    // ---------- one-hot column info for n = n0 + rlane ----------
    const int  ncol   = n0 + rlane;
    const bool nvalid = (ncol < N_DIM);
    const int  nld    = nvalid ? ncol : (N_DIM - 1);   // clamped load index
    int  gcls  = gt_cls[(size_t)b * N_DIM + nld];
    bool gvald = gt_valid[(size_t)b * N_DIM + nld] != 0;
    if (!nvalid) { gcls = -1; gvald = false; }

    // ---------- cls_cost GEMM: probs[16xC] @ onehot[Cx16] via f32 WMMA ----------
    // A 16x4 layout: lanes 0-15 hold (M=rlane, K=4k+{0,1}); lanes 16-31 K=4k+{2,3}
    // B 4x16 layout: lanes 0-15 hold (N=rlane, K=4k+{0,1}); lanes 16-31 K=4k+{2,3}
    v8f acc = {};
    for (int k = 0; k < KSTEPS; ++k) {
        const int kk = k * 4 + hi * 2;
        v2f a, bm;
        a.x = __expf(logits[kk]     - m) * inv_s;
        a.y = __expf(logits[kk + 1] - m) * inv_s;
        bm.x = (gcls == kk)     ? 1.0f : 0.0f;
        bm.y = (gcls == kk + 1) ? 1.0f : 0.0f;
        acc = __builtin_amdgcn_wmma_f32_16x16x4_f32(
            /*neg_a=*/false, a, /*neg_b=*/false, bm,
            /*c_mod=*/(short)0, acc, /*reuse_a=*/false, /*reuse_b=*/false);
    }
    // acc[v] = probs[q0 + v + hi*8][ gt_cls[ncol] ]  at column ncol

    // ---------- epilogue: GIoU + L2 per element, fuse and store ----------
    const float* gb = gt_boxes + ((size_t)b * N_DIM + nld) * 4;
    const float gx0 = gb[0], gy0 = gb[1], gx1 = gb[2], gy1 = gb[3];
    const float area_g = (gx1 - gx0) * (gy1 - gy0);

    float* outb = out + (size_t)b * Q_DIM * N_DIM;

#pragma unroll
    for (int v = 0; v < 8; ++v) {
        const int q = q0 + v + hi * 8;
        if (q >= Q_DIM) continue;                  // rows 900..911 of last tile
        const float* pb = pred_boxes + ((size_t)b * Q_DIM + q) * 4;
        const float px0 = pb[0], py0 = pb[1], px1 = pb[2], py1 = pb[3];

        const float area_p = (px1 - px0) * (py1 - py0);
        const float iw = fmaxf(fminf(px1, gx1) - fmaxf(px0, gx0), 0.0f);
        const float ih = fmaxf(fminf(py1, gy1) - fmaxf(py0, gy0), 0.0f);
        const float inter = iw * ih;
        const float uni   = area_p + area_g - inter;
        const float iou   = inter / uni;

        const float ew = fmaxf(fmaxf(px1, gx1) - fminf(px0, gx0), 0.0f);
        const float eh = fmaxf(fmaxf(py1, gy1) - fminf(py0, gy0), 0.0f);
        const float enc = ew * eh;
        const float giou = iou - (enc - uni) / enc;

        const float dx0 = px0 - gx0, dy0 = py0 - gy0;
        const float dx1 = px1 - gx1, dy1 = py1 - gy1;
        const float l1 = sqrtf(dx0 * dx0 + dy0 * dy0 + dx1 * dx1 + dy1 * dy1);

        float cost = -acc[v] + GIOU_W * (1.0f - giou) + L1_W * l1;
        cost = gvald ? cost : INVALID_COST;

        if (nvalid) outb[(size_t)q * N_DIM + ncol] = cost;
    }
}

extern "C" void kernel_launch(void* const* d_in, const int* in_sizes, int n_in,
                              void* d_out, int out_size, void* d_ws, size_t ws_size,
                              hipStream_t stream) {
    const float* pred_boxes = (const float*)d_in[0];
    const float* pred_cls   = (const float*)d_in[1];
    const float* gt_boxes   = (const float*)d_in[2];
    const int*   gt_cls     = (const int*)d_in[3];
    const unsigned char* gt_valid = (const unsigned char*)d_in[4];
    float* out = (float*)d_out;

    dim3 grid((Q_DIM + 15) / 16, (N_DIM + 15) / 16, B_DIM);  // 57 x 19 x 128
    detr_cost_wmma<<<grid, 32, 0, stream>>>(pred_boxes, pred_cls, gt_boxes,
                                            gt_cls, gt_valid, out);
    (void)in_sizes; (void)n_in; (void)out_size; (void)d_ws; (void)ws_size;
}